// EnergyModel_4114578670361
// MI455X (gfx1250) — compile-verified
//
#include <hip/hip_runtime.h>
#include <math.h>

typedef __attribute__((ext_vector_type(2))) float v2f;
typedef __attribute__((ext_vector_type(8))) float v8f;

#define NQ        128
#define FDIM      576
#define NGRP      192
#define F4_PER_Q  (FDIM / 4)        // 144
#define F4_PER_T  (NQ * F4_PER_Q)   // 18432 float4 per T-row per tensor
#define BLOCK     256
#define LN2F      0.69314718055994531f

__global__ __launch_bounds__(BLOCK) void energy_kernel(
    const float* __restrict__ T,
    const float* __restrict__ desc,
    const float* __restrict__ qf,
    const float* __restrict__ qatt,
    const float* __restrict__ logit,
    const float* __restrict__ ranges,
    float* __restrict__ out)
{
    __shared__ float4 s_scale4[F4_PER_Q];   // per-feature weight, 4 at a time
    __shared__ float  s_att[NQ];
    __shared__ float  s_part[BLOCK / 32];

    const int tid = threadIdx.x;
    const int t   = blockIdx.x;

    // ---- build per-feature scale = 2*softplus(logit[g(f)]) / (LN2*NGRP) ----
    if (tid < F4_PER_Q) {
        float4 sc;
        float* scp = (float*)&sc;
        #pragma unroll
        for (int j = 0; j < 4; ++j) {
            int f = tid * 4 + j;
            int g = (f < 64) ? f
                  : (f < 256) ? (64  + (f - 64)  / 3)
                              : (128 + (f - 256) / 5);
            float x  = logit[g];
            float sp = (x > 20.0f) ? x : log1pf(expf(x));
            scp[j] = 2.0f * sp / (LN2F * (float)NGRP);
        }
        s_scale4[tid] = sc;
    }
    if (tid < NQ) s_att[tid] = qatt[tid];
    __syncthreads();

    // ---- streaming weighted sum of squared differences (HBM-bound) ----
    const float4* dp = (const float4*)desc + (size_t)t * F4_PER_T;
    const float4* qp = (const float4*)qf   + (size_t)t * F4_PER_T;

    float acc0 = 0.0f, acc1 = 0.0f;     // two accumulators -> A-matrix K=0/K=1
    #pragma unroll 4
    for (int e = tid; e < F4_PER_T; e += BLOCK) {
        int q = e / F4_PER_Q;           // attention index
        int k = e - q * F4_PER_Q;       // float4 index within feature row
        float4 d = dp[e];
        float4 v = qp[e];
        float4 s = s_scale4[k];
        float  a = s_att[q];
        float dx = d.x - v.x, dy = d.y - v.y;
        float dz = d.z - v.z, dw = d.w - v.w;
        acc0 += a * (dx * dx * s.x + dy * dy * s.y);
        acc1 += a * (dz * dz * s.z + dw * dw * s.w);
    }

    // ---- exact f32 wave reduction via V_WMMA_F32_16X16X4_F32 ----
    // A (16x4): lane L<16 supplies (M=L, K=0)=acc0, (M=L, K=1)=acc1;
    //           lane L>=16 supplies (M=L-16, K=2)=acc0, (K=3)=acc1.
    // B = all ones  =>  D[m,n] = acc0[m]+acc1[m]+acc0[m+16]+acc1[m+16]
    // (independent of B's lane layout).
    v2f A; A.x = acc0; A.y = acc1;
    v2f B; B.x = 1.0f; B.y = 1.0f;
    v8f C = {};
    C = __builtin_amdgcn_wmma_f32_16x16x4_f32(
            /*neg_a=*/false, A, /*neg_b=*/false, B,
            /*c_mod=*/(short)0, C, /*reuse_a=*/false, /*reuse_b=*/false);
    // lane n<16 holds D[0..7,n]; lane n>=16 holds D[8..15,n-16]
    float s = C[0] + C[1] + C[2] + C[3] + C[4] + C[5] + C[6] + C[7];
    s += __shfl_xor(s, 16, 32);         // rows 0-7 total + rows 8-15 total

    const int lane = tid & 31;
    const int wid  = tid >> 5;
    if (lane == 0) s_part[wid] = s;
    __syncthreads();

    if (tid == 0) {
        float e = 0.0f;
        #pragma unroll
        for (int w = 0; w < BLOCK / 32; ++w) e += s_part[w];
        // range mask on T[t, 4:7]
        bool ok = true;
        #pragma unroll
        for (int i = 0; i < 3; ++i) {
            float x = T[t * 7 + 4 + i];
            ok = ok && (ranges[2 * i + 1] >= x) && (x >= ranges[2 * i]);
        }
        out[t] = ok ? e : 100000.0f;
    }
}

extern "C" void kernel_launch(void* const* d_in, const int* in_sizes, int n_in,
                              void* d_out, int out_size, void* d_ws, size_t ws_size,
                              hipStream_t stream) {
    const float* T      = (const float*)d_in[0];
    const float* desc   = (const float*)d_in[1];
    const float* qf     = (const float*)d_in[2];
    const float* qatt   = (const float*)d_in[3];
    const float* logit  = (const float*)d_in[4];
    const float* ranges = (const float*)d_in[5];
    float* out = (float*)d_out;

    const int Nt = in_sizes[0] / 7;     // T is (Nt, 7)
    energy_kernel<<<Nt, BLOCK, 0, stream>>>(T, desc, qf, qatt, logit, ranges, out);
}